// Attention_68805376082373
// MI455X (gfx1250) — compile-verified
//
#include <hip/hip_runtime.h>

// Problem sizes
constexpr int BB  = 32;
constexpr int NN  = 2048;
constexpr int LL  = 1454;
constexpr int DD  = 256;
constexpr int HC1 = 727;
constexpr int HC2 = 363;

typedef __attribute__((ext_vector_type(2))) float v2f;
typedef __attribute__((ext_vector_type(8))) float v8f;

constexpr int KC   = 32;        // K-chunk of L per LDS stage
constexpr int HSTR = KC + 2;    // padded LDS row stride (34 -> conflict-free banks)

// ---------------------------------------------------------------------------
// Kernel 1: attention scores  s[b,n] = tanh(H[b,n,:]@W1 + b1) @ W2 + b2
//           fused with rowsum (padding mask).  16 rows per WG, 8 waves,
//           each wave owns 32 of the 256 D-columns via v_wmma_f32_16x16x4_f32.
// ---------------------------------------------------------------------------
__global__ __launch_bounds__(256)
void attn_scores_kernel(const float* __restrict__ Hm, const float* __restrict__ W1,
                        const float* __restrict__ b1, const float* __restrict__ W2,
                        const float* __restrict__ b2,
                        float* __restrict__ scoresRaw, float* __restrict__ rowsumOut)
{
    __shared__ float Ht[16 * HSTR];      // 16 x KC tile of H (padded)
    __shared__ float W1t[KC * DD];       // KC x 256 tile of W1
    __shared__ float score16[16];
    __shared__ float rowsum16[16];

    const int t    = threadIdx.x;
    const long row0 = (long)blockIdx.x * 16;   // first row of flattened [B*N]

    const int w    = t >> 5;       // wave id 0..7
    const int lane = t & 31;
    const int half = lane >> 4;    // 0: lanes 0-15, 1: lanes 16-31
    const int ln   = lane & 15;

    const int lr = t >> 4;         // staging: row 0..15
    const int lc = t & 15;         // staging: col group

    const int col0 = w * 32 + ln;  // this lane's first D-column

    float rs = 0.0f;               // running row-sum for row lr
    v8f acc0 = {};                 // C tile: rows 0..15 x cols [col0]
    v8f acc1 = {};                 // C tile: rows 0..15 x cols [col0+16]

    for (int kc = 0; kc < LL; kc += KC) {
        // --- stage H tile (coalesced: 16 consecutive lanes read 64B of a row)
        #pragma unroll
        for (int i = 0; i < KC / 16; ++i) {
            int c = lc + i * 16;
            float v = 0.0f;
            if (kc + c < LL) v = Hm[(row0 + lr) * LL + kc + c];
            Ht[lr * HSTR + c] = v;
            rs += v;
        }
        // --- stage W1 tile (each iter: 256 lanes read one full W1 row)
        #pragma unroll 4
        for (int i = 0; i < KC; ++i) {
            float v = 0.0f;
            if (kc + i < LL) v = W1[(long)(kc + i) * DD + t];
            W1t[i * DD + t] = v;
        }
        __syncthreads();

        // --- WMMA over this K-chunk (f32 16x16x4)
        #pragma unroll
        for (int k = 0; k < KC; k += 4) {
            // A 16x4: lanes 0-15 hold K=k..k+1, lanes 16-31 hold K=k+2..k+3
            v2f a;
            a.x = Ht[ln * HSTR + k + 2 * half];
            a.y = Ht[ln * HSTR + k + 2 * half + 1];
            // B 4x16: same K-split across lane halves, N striped over lanes
            v2f bA, bB;
            bA.x = W1t[(k + 2 * half)     * DD + col0];
            bA.y = W1t[(k + 2 * half + 1) * DD + col0];
            bB.x = W1t[(k + 2 * half)     * DD + col0 + 16];
            bB.y = W1t[(k + 2 * half + 1) * DD + col0 + 16];
            acc0 = __builtin_amdgcn_wmma_f32_16x16x4_f32(false, a, false, bA,
                                                         (short)0, acc0, false, false);
            acc1 = __builtin_amdgcn_wmma_f32_16x16x4_f32(false, a, false, bB,
                                                         (short)0, acc1, false, false);
        }
        __syncthreads();
    }

    // --- reduce row-sums: 16 lanes per half-wave hold the same row (lr = 2w+half)
    #pragma unroll
    for (int o = 1; o < 16; o <<= 1) rs += __shfl_xor(rs, o, 32);
    if (ln == 0) rowsum16[2 * w + half] = rs;
    if (t < 16)  score16[t] = b2[0];
    __syncthreads();

    // --- epilogue: tanh(x + b1) * W2, reduce over the 256 D-columns
    // C layout: acc[j] holds row (j + 8*half), col (base + ln)
    float p[8];
    #pragma unroll
    for (int j = 0; j < 8; ++j) {
        float xa = acc0[j] + b1[col0];
        float xb = acc1[j] + b1[col0 + 16];
        p[j] = tanhf(xa) * W2[col0] + tanhf(xb) * W2[col0 + 16];
    }
    #pragma unroll
    for (int o = 1; o < 16; o <<= 1) {
        #pragma unroll
        for (int j = 0; j < 8; ++j) p[j] += __shfl_xor(p[j], o, 32);
    }
    if (ln == 0) {
        #pragma unroll
        for (int j = 0; j < 8; ++j) atomicAdd(&score16[8 * half + j], p[j]);
    }
    __syncthreads();

    if (t < 16) {
        float sum = rowsum16[t];
        float sc  = score16[t];
        scoresRaw[row0 + t] = (sum == 0.0f) ? 0.0f : sc;   // zero BEFORE softmax
        rowsumOut[row0 + t] = sum;
    }
}

// ---------------------------------------------------------------------------
// Kernel 2: masked softmax over N with faithful sum(A + 1e-9) renorm.
//           Also zero-inits this batch's M row in d_out (atomics target).
// ---------------------------------------------------------------------------
__global__ __launch_bounds__(256)
void softmax_kernel(const float* __restrict__ scoresRaw, const float* __restrict__ rowsum,
                    float* __restrict__ Aout, float* __restrict__ Mout)
{
    __shared__ float red[256];
    __shared__ float sA[NN];
    const int t = threadIdx.x;
    const int b = blockIdx.x;

    float mx = -3.402823466e+38f;
    for (int j = 0; j < NN / 256; ++j) {
        float v = scoresRaw[b * NN + t + j * 256];
        sA[t + j * 256] = v;
        mx = fmaxf(mx, v);
    }
    red[t] = mx; __syncthreads();
    for (int s = 128; s > 0; s >>= 1) {
        if (t < s) red[t] = fmaxf(red[t], red[t + s]);
        __syncthreads();
    }
    mx = red[0]; __syncthreads();

    float zs = 0.0f;
    for (int j = 0; j < NN / 256; ++j) zs += expf(sA[t + j * 256] - mx);
    red[t] = zs; __syncthreads();
    for (int s = 128; s > 0; s >>= 1) {
        if (t < s) red[t] += red[t + s];
        __syncthreads();
    }
    float Z = red[0]; __syncthreads();

    float ds = 0.0f;
    for (int j = 0; j < NN / 256; ++j) {
        int i = t + j * 256;
        float pv = expf(sA[i] - mx) / Z;
        if (rowsum[b * NN + i] == 0.0f) pv = 0.0f;   // zero AFTER softmax
        sA[i] = pv;
        ds += pv + 1e-9f;                             // epsilon for ALL entries
    }
    red[t] = ds; __syncthreads();
    for (int s = 128; s > 0; s >>= 1) {
        if (t < s) red[t] += red[t + s];
        __syncthreads();
    }
    float denom = red[0]; __syncthreads();

    for (int j = 0; j < NN / 256; ++j) {
        int i = t + j * 256;
        Aout[b * NN + i] = sA[i] / denom;
    }
    for (int l = t; l < LL; l += 256) Mout[b * LL + l] = 0.0f;
}

// ---------------------------------------------------------------------------
// Kernel 3: attention pooling M[b,l] = sum_n A[b,n] * H[b,n,l]
//           (second, final pass over H; 16 N-chunks per batch for parallelism)
// ---------------------------------------------------------------------------
constexpr int NCHUNKS = 16;
constexpr int CHN     = NN / NCHUNKS;   // 128

__global__ __launch_bounds__(256)
void pool_kernel(const float* __restrict__ Hm, const float* __restrict__ Aout,
                 float* __restrict__ Mout)
{
    __shared__ float aT[CHN];
    const int t  = threadIdx.x;
    const int b  = blockIdx.x / NCHUNKS;
    const int ch = blockIdx.x % NCHUNKS;
    const int n0 = ch * CHN;

    if (t < CHN) aT[t] = Aout[b * NN + n0 + t];
    __syncthreads();

    const float* Hb = Hm + ((long)b * NN + n0) * LL;
    for (int cb = 0; cb < LL; cb += 256) {
        int col = cb + t;
        if (col < LL) {
            float acc = 0.0f;
            #pragma unroll 4
            for (int nn = 0; nn < CHN; ++nn)
                acc += aT[nn] * Hb[(long)nn * LL + col];
            atomicAdd(&Mout[b * LL + col], acc);
        }
    }
}

// ---------------------------------------------------------------------------
// Kernel 4: classifier  relu(M@Wc1+bc1) -> relu(@Wc2+bc2) -> softmax(@Wc3+bc3)
// ---------------------------------------------------------------------------
__global__ __launch_bounds__(256)
void classifier_kernel(const float* __restrict__ Mout,
                       const float* __restrict__ Wc1, const float* __restrict__ bc1,
                       const float* __restrict__ Wc2, const float* __restrict__ bc2,
                       const float* __restrict__ Wc3, const float* __restrict__ bc3,
                       float* __restrict__ Y)
{
    __shared__ float Mb[LL];
    __shared__ float h1s[HC1];
    __shared__ float h2s[HC2];
    __shared__ float lg[2];
    const int t = threadIdx.x;
    const int b = blockIdx.x;

    for (int l = t; l < LL; l += 256) Mb[l] = Mout[b * LL + l];
    __syncthreads();

    for (int j = t; j < HC1; j += 256) {
        float acc = bc1[j];
        for (int l = 0; l < LL; ++l) acc += Mb[l] * Wc1[l * HC1 + j];
        h1s[j] = fmaxf(acc, 0.0f);
    }
    __syncthreads();

    for (int j = t; j < HC2; j += 256) {
        float acc = bc2[j];
        for (int l = 0; l < HC1; ++l) acc += h1s[l] * Wc2[l * HC2 + j];
        h2s[j] = fmaxf(acc, 0.0f);
    }
    __syncthreads();

    if (t < 2) {
        float acc = bc3[t];
        for (int l = 0; l < HC2; ++l) acc += h2s[l] * Wc3[l * 2 + t];
        lg[t] = acc;
    }
    __syncthreads();

    if (t == 0) {
        float m  = fmaxf(lg[0], lg[1]);
        float e0 = expf(lg[0] - m), e1 = expf(lg[1] - m);
        float s  = e0 + e1;
        Y[b * 2 + 0] = e0 / s;
        Y[b * 2 + 1] = e1 / s;
    }
}

// ---------------------------------------------------------------------------
extern "C" void kernel_launch(void* const* d_in, const int* in_sizes, int n_in,
                              void* d_out, int out_size, void* d_ws, size_t ws_size,
                              hipStream_t stream)
{
    const float* Hm  = (const float*)d_in[0];
    const float* W1  = (const float*)d_in[1];
    const float* b1  = (const float*)d_in[2];
    const float* W2  = (const float*)d_in[3];
    const float* b2  = (const float*)d_in[4];
    const float* Wc1 = (const float*)d_in[5];
    const float* bc1 = (const float*)d_in[6];
    const float* Wc2 = (const float*)d_in[7];
    const float* bc2 = (const float*)d_in[8];
    const float* Wc3 = (const float*)d_in[9];
    const float* bc3 = (const float*)d_in[10];

    float* out = (float*)d_out;
    float* Y = out;                        // [32, 2]
    float* M = out + BB * 2;               // [32, 1454]
    float* A = out + BB * 2 + BB * LL;     // [32, 1, 2048]

    float* ws        = (float*)d_ws;
    float* scoresRaw = ws;                 // [B*N]
    float* rowsum    = ws + BB * NN;       // [B*N]

    attn_scores_kernel<<<(BB * NN) / 16, 256, 0, stream>>>(Hm, W1, b1, W2, b2,
                                                           scoresRaw, rowsum);
    softmax_kernel<<<BB, 256, 0, stream>>>(scoresRaw, rowsum, A, M);
    pool_kernel<<<BB * NCHUNKS, 256, 0, stream>>>(Hm, A, M);
    classifier_kernel<<<BB, 256, 0, stream>>>(M, Wc1, bc1, Wc2, bc2, Wc3, bc3, Y);
}